// TGCN_PyG_76012331204905
// MI455X (gfx1250) — compile-verified
//
#include <hip/hip_runtime.h>
#include <hip/hip_bf16.h>
#include <math.h>

typedef __attribute__((ext_vector_type(2))) float v2f;
typedef __attribute__((ext_vector_type(8))) float v8f;

#define FDIM 64
#define TSTEPS 12

__device__ __forceinline__ v8f wmma4(v2f a, v2f b, v8f c) {
    // D = A(16x4,f32) * B(4x16,f32) + C(16x16,f32)
    return __builtin_amdgcn_wmma_f32_16x16x4_f32(false, a, false, b, (short)0, c, false, false);
}

__device__ __forceinline__ float sigmoidf_(float x) {
    return 1.0f / (1.0f + __expf(-x));
}

// ---------------- setup kernels ----------------

__global__ void init_kernel(float* __restrict__ deg, float* __restrict__ h, int N) {
    int i = blockIdx.x * blockDim.x + threadIdx.x;
    if (i < N * FDIM) h[i] = 0.0f;
    if (i < N) deg[i] = 1.0f;           // self-loop contributes 1 to degree
}

__global__ void deg_kernel(const int* __restrict__ col, float* __restrict__ deg, int E) {
    int e = blockIdx.x * blockDim.x + threadIdx.x;
    if (e < E) atomicAdd(deg + col[e], 1.0f);
}

__global__ void norm_kernel(const int* __restrict__ row, const int* __restrict__ col,
                            const float* __restrict__ deg, float* __restrict__ norm, int E) {
    int e = blockIdx.x * blockDim.x + threadIdx.x;
    if (e < E) norm[e] = rsqrtf(deg[row[e]]) * rsqrtf(deg[col[e]]);
}

// ---------------- GCN linear: xl = x_t @ W_gcn^T + b_gcn  (WMMA f32) ----------------
// One wave per 16-row strip; 4 output column tiles of 16; K=64 in 16 chunks of 4.

__global__ void gcn_linear_kernel(const float* __restrict__ x, const float* __restrict__ Wg,
                                  const float* __restrict__ bg, float* __restrict__ xl, int N) {
    int wave  = (blockIdx.x * blockDim.x + threadIdx.x) >> 5;
    int lane  = threadIdx.x & 31;
    int rbase = wave * 16;
    if (rbase >= N) return;                 // uniform across wave
    const bool full = (rbase + 16 <= N);    // uniform: full-tile fast path
    int mlane = lane & 15;
    int hi    = lane >> 4;
    int khalf = hi << 1;                    // 0 or 2

    int arow_idx = rbase + mlane; if (arow_idx >= N) arow_idx = N - 1;
    const float* arow = x + (size_t)arow_idx * FDIM;

    v2f afr[16];
#pragma unroll
    for (int kc = 0; kc < 16; ++kc)
        afr[kc] = *(const v2f*)(arow + kc * 4 + khalf);

#pragma unroll
    for (int ct = 0; ct < 4; ++ct) {
        const float* brow = Wg + (size_t)(ct * 16 + mlane) * FDIM;   // B[k,n] = Wg[n,k]
        v8f acc = {};
#pragma unroll
        for (int kc = 0; kc < 16; ++kc) {
            v2f bfr = *(const v2f*)(brow + kc * 4 + khalf);
            acc = wmma4(afr[kc], bfr, acc);
        }
        float bias = bg[ct * 16 + mlane];
        float* outp = xl + (size_t)(rbase + 8 * hi) * FDIM + ct * 16 + mlane;
        if (full) {
#pragma unroll
            for (int v = 0; v < 8; ++v)
                outp[(size_t)v * FDIM] = acc[v] + bias;
        } else {
#pragma unroll
            for (int v = 0; v < 8; ++v) {
                int r = rbase + v + 8 * hi;
                if (r < N) xl[(size_t)r * FDIM + ct * 16 + mlane] = acc[v] + bias;
            }
        }
    }
}

// ---------------- self-loop seed: agg = (1/deg[n]) * xl ----------------

__global__ void agginit_kernel(const float* __restrict__ xl, const float* __restrict__ deg,
                               float* __restrict__ agg, int N) {
    int i = blockIdx.x * blockDim.x + threadIdx.x;
    if (i < N * FDIM) {
        float d = deg[i >> 6];
        agg[i] = xl[i] * (1.0f / d);       // dinv[n]^2 = 1/deg[n]
    }
}

// ---------------- edge scatter: agg[col] += norm * xl[row] ----------------

__global__ void scatter_kernel(const int* __restrict__ row, const int* __restrict__ col,
                               const float* __restrict__ norm, const float* __restrict__ xl,
                               float* __restrict__ agg, long long EF) {
    long long t = (long long)blockIdx.x * blockDim.x + threadIdx.x;
    if (t >= EF) return;
    int e = (int)(t >> 6);
    int f = (int)(t & 63);
    int r = row[e], c = col[e];
    float v = norm[e] * xl[(size_t)r * FDIM + f];
    atomicAdd(agg + (size_t)c * FDIM + f, v);
}

// ---------------- fused GRU: gi = agg@Wih^T+bih, gh = h@Whh^T+bhh, gates, h inplace ----------------
// One wave per 16-row strip. A-fragments of agg and h kept in registers; per gate-column
// tile (4 of them) we accumulate 6 WMMA tiles (i_r,i_z,i_n,h_r,h_z,h_n) and do the
// elementwise gate math directly in C-layout registers.

__global__ void gru_kernel(const float* __restrict__ agg, float* __restrict__ h,
                           const float* __restrict__ Wih, const float* __restrict__ Whh,
                           const float* __restrict__ bih, const float* __restrict__ bhh, int N) {
    int wave  = (blockIdx.x * blockDim.x + threadIdx.x) >> 5;
    int lane  = threadIdx.x & 31;
    int rbase = wave * 16;
    if (rbase >= N) return;
    const bool full = (rbase + 16 <= N);
    int mlane = lane & 15;
    int hi    = lane >> 4;
    int khalf = hi << 1;

    int arow_idx = rbase + mlane; if (arow_idx >= N) arow_idx = N - 1;
    const float* arow_a = agg + (size_t)arow_idx * FDIM;
    const float* arow_h = h   + (size_t)arow_idx * FDIM;

    v2f afa[16], afh[16];
#pragma unroll
    for (int kc = 0; kc < 16; ++kc) {
        afa[kc] = *(const v2f*)(arow_a + kc * 4 + khalf);
        afh[kc] = *(const v2f*)(arow_h + kc * 4 + khalf);
    }

#pragma unroll
    for (int ct = 0; ct < 4; ++ct) {
        int g_r = 0   + ct * 16 + mlane;   // gate output column indices (r, z, n blocks)
        int g_z = 64  + ct * 16 + mlane;
        int g_n = 128 + ct * 16 + mlane;
        const float* b_ir = Wih + (size_t)g_r * FDIM;
        const float* b_iz = Wih + (size_t)g_z * FDIM;
        const float* b_in = Wih + (size_t)g_n * FDIM;
        const float* b_hr = Whh + (size_t)g_r * FDIM;
        const float* b_hz = Whh + (size_t)g_z * FDIM;
        const float* b_hn = Whh + (size_t)g_n * FDIM;

        v8f ir = {}, iz = {}, inn = {}, hr = {}, hz = {}, hn = {};
#pragma unroll
        for (int kc = 0; kc < 16; ++kc) {
            int ko = kc * 4 + khalf;
            ir  = wmma4(afa[kc], *(const v2f*)(b_ir + ko), ir);
            iz  = wmma4(afa[kc], *(const v2f*)(b_iz + ko), iz);
            inn = wmma4(afa[kc], *(const v2f*)(b_in + ko), inn);
            hr  = wmma4(afh[kc], *(const v2f*)(b_hr + ko), hr);
            hz  = wmma4(afh[kc], *(const v2f*)(b_hz + ko), hz);
            hn  = wmma4(afh[kc], *(const v2f*)(b_hn + ko), hn);
        }
        float bir = bih[g_r], biz = bih[g_z], bin = bih[g_n];
        float bhr = bhh[g_r], bhz = bhh[g_z], bhn = bhh[g_n];

        float* hp = h + (size_t)(rbase + 8 * hi) * FDIM + ct * 16 + mlane;
        if (full) {
#pragma unroll
            for (int v = 0; v < 8; ++v) {
                float ho = hp[(size_t)v * FDIM];
                float rg = sigmoidf_((ir[v] + bir) + (hr[v] + bhr));
                float zg = sigmoidf_((iz[v] + biz) + (hz[v] + bhz));
                float ng = tanhf((inn[v] + bin) + rg * (hn[v] + bhn));
                hp[(size_t)v * FDIM] = (1.0f - zg) * ng + zg * ho;
            }
        } else {
#pragma unroll
            for (int v = 0; v < 8; ++v) {
                int r = rbase + v + 8 * hi;
                if (r >= N) continue;
                size_t hidx = (size_t)r * FDIM + ct * 16 + mlane;
                float ho = h[hidx];
                float rg = sigmoidf_((ir[v] + bir) + (hr[v] + bhr));
                float zg = sigmoidf_((iz[v] + biz) + (hz[v] + bhz));
                float ng = tanhf((inn[v] + bin) + rg * (hn[v] + bhn));
                h[hidx] = (1.0f - zg) * ng + zg * ho;
            }
        }
    }
}

// ---------------- output head: out = h @ W_out^T + b_out ----------------

__global__ void out_kernel(const float* __restrict__ h, const float* __restrict__ Wout,
                           const float* __restrict__ bout, float* __restrict__ out, int N) {
    int n = blockIdx.x * blockDim.x + threadIdx.x;
    if (n >= N) return;
    const float* hrow = h + (size_t)n * FDIM;
    float s = 0.0f;
#pragma unroll
    for (int k = 0; k < FDIM; ++k) s += hrow[k] * Wout[k];
    out[n] = s + bout[0];
}

// ---------------- launcher ----------------

extern "C" void kernel_launch(void* const* d_in, const int* in_sizes, int n_in,
                              void* d_out, int out_size, void* d_ws, size_t ws_size,
                              hipStream_t stream) {
    const float* x_seq = (const float*)d_in[0];
    const int*   ei    = (const int*)d_in[1];
    const float* Wg    = (const float*)d_in[2];
    const float* bg    = (const float*)d_in[3];
    const float* Wih   = (const float*)d_in[4];
    const float* Whh   = (const float*)d_in[5];
    const float* bih   = (const float*)d_in[6];
    const float* bhh   = (const float*)d_in[7];
    const float* Wout  = (const float*)d_in[8];
    const float* bout  = (const float*)d_in[9];
    float* out = (float*)d_out;

    const int E = in_sizes[1] / 2;                 // edge_index is (2, E)
    const int N = in_sizes[0] / (TSTEPS * FDIM);   // x_seq is (T, N, F)

    // workspace layout
    float* deg  = (float*)d_ws;                    // N
    float* norm = deg + N;                         // E
    float* xl   = norm + E;                        // N*64
    float* agg  = xl + (size_t)N * FDIM;           // N*64
    float* h    = agg + (size_t)N * FDIM;          // N*64

    const int* row = ei;
    const int* col = ei + E;

    const int TB = 256;
    int nf_blocks    = (N * FDIM + TB - 1) / TB;
    int e_blocks     = (E + TB - 1) / TB;
    int strip_blocks = ((N + 15) / 16 + 7) / 8;    // 8 waves per 256-thread block
    long long EF     = (long long)E * FDIM;
    int ef_blocks    = (int)((EF + TB - 1) / TB);

    init_kernel<<<nf_blocks, TB, 0, stream>>>(deg, h, N);
    deg_kernel<<<e_blocks, TB, 0, stream>>>(col, deg, E);
    norm_kernel<<<e_blocks, TB, 0, stream>>>(row, col, deg, norm, E);

    for (int t = 0; t < TSTEPS; ++t) {
        const float* x_t = x_seq + (size_t)t * N * FDIM;
        gcn_linear_kernel<<<strip_blocks, TB, 0, stream>>>(x_t, Wg, bg, xl, N);
        agginit_kernel<<<nf_blocks, TB, 0, stream>>>(xl, deg, agg, N);
        scatter_kernel<<<ef_blocks, TB, 0, stream>>>(row, col, norm, xl, agg, EF);
        gru_kernel<<<strip_blocks, TB, 0, stream>>>(agg, h, Wih, Whh, bih, bhh, N);
    }

    out_kernel<<<(N + TB - 1) / TB, TB, 0, stream>>>(h, Wout, bout, out, N);
}